// SSMClassifier_44667659878984
// MI455X (gfx1250) — compile-verified
//
#include <hip/hip_runtime.h>
#include <math.h>
#include <stdint.h>

typedef __attribute__((ext_vector_type(2))) float v2f;
typedef __attribute__((ext_vector_type(8))) float v8f;
typedef __attribute__((__vector_size__(4 * sizeof(int)))) int v4i;

#define Bb 64
#define Ll 4096
#define Ff 28
#define Fp 32
#define Dd 128
#define Ss 64
#define Cc 10
#define TC 64                 // scan chunk length
#define NCH (Ll / TC)         // 64 chunks
#define LPB 32                // l-steps per block in fused output kernel

#if defined(__has_builtin)
#if __has_builtin(__builtin_amdgcn_global_load_async_to_lds_b128)
#define HAVE_ASYNC_CP 1
#endif
#if __has_builtin(__builtin_amdgcn_s_wait_asynccnt)
#define HAVE_WAIT_ASYNC 1
#endif
#endif

// Async 16B global->LDS copy (CDNA5 GLOBAL_LOAD_ASYNC_TO_LDS_B128, ASYNCcnt).
// Generic LDS pointers carry the LDS byte offset in their low 32 bits, so the
// integer round-trip yields a valid 32-bit AS(3) pointer.
__device__ __forceinline__ void async_cp16(const float* g, float* l) {
#ifdef HAVE_ASYNC_CP
  __builtin_amdgcn_global_load_async_to_lds_b128(
      (__attribute__((address_space(1))) v4i*)(uintptr_t)g,
      (__attribute__((address_space(3))) v4i*)(uint32_t)(uintptr_t)l, 0, 0);
#else
  *(float4*)l = *(const float4*)g;
#endif
}

__device__ __forceinline__ void wait_async0() {
#ifdef HAVE_ASYNC_CP
#ifdef HAVE_WAIT_ASYNC
  __builtin_amdgcn_s_wait_asynccnt(0);
#else
  asm volatile("s_wait_asynccnt 0x0" ::: "memory");
#endif
#endif
}

// ---- CDNA5 fp32 WMMA: D = A(16x4) * B(4x16) + C(16x16) ----
__device__ __forceinline__ v8f wmma4(v2f a, v2f b, v8f c) {
  return __builtin_amdgcn_wmma_f32_16x16x4_f32(false, a, false, b, (short)0, c,
                                               false, false);
}

// A-frag (16x4 f32): lanes 0-15 rows M, VGPR0 K=k0+2h, VGPR1 K=k0+2h+1
__device__ __forceinline__ v2f ldA(const float* m, int ld, int row0, int k0, int lane) {
  int r = lane & 15, h = lane >> 4;
  v2f a;
  a[0] = m[(row0 + r) * ld + k0 + 2 * h + 0];
  a[1] = m[(row0 + r) * ld + k0 + 2 * h + 1];
  return a;
}
// B-frag (4x16 f32): lanes 0-15 cols N, VGPR v holds K=k0+v+2h
__device__ __forceinline__ v2f ldB(const float* m, int ld, int k0, int col0, int lane) {
  int r = lane & 15, h = lane >> 4;
  v2f b;
  b[0] = m[(k0 + 2 * h + 0) * ld + col0 + r];
  b[1] = m[(k0 + 2 * h + 1) * ld + col0 + r];
  return b;
}
// C/D element (v, lane) lives at (row0 + v + 8h, col0 + r)

__device__ __forceinline__ float gelu_exact(float x) {
  return 0.5f * x * (1.0f + erff(x * 0.70710678118654752440f));
}

// ---------------- K1: tiny prep (M = W_in^T Bm^T, cvec, AT powers) ----------
__global__ __launch_bounds__(256) void k1_prep(const float* __restrict__ W_in,
                                               const float* __restrict__ b_in,
                                               const float* __restrict__ A,
                                               const float* __restrict__ Bm,
                                               float* __restrict__ M,
                                               float* __restrict__ cvec,
                                               float* __restrict__ Apow) {
  __shared__ __align__(16) float ATl[Ss * Ss];
  __shared__ __align__(16) float pa[Ss * Ss];
  __shared__ __align__(16) float pb[Ss * Ss];
  int tid = threadIdx.x;
  // M[f][s] = sum_d W_in[d][f] * Bm[s][d]  (F padded 28 -> 32 with zeros)
  for (int idx = tid; idx < Fp * Ss; idx += 256) {
    int f = idx / Ss, s = idx % Ss;
    float acc = 0.f;
    if (f < Ff)
      for (int d = 0; d < Dd; ++d) acc += W_in[d * Ff + f] * Bm[s * Dd + d];
    M[idx] = acc;
  }
  // cvec[s] = sum_d b_in[d] * Bm[s][d]
  for (int s = tid; s < Ss; s += 256) {
    float acc = 0.f;
    for (int d = 0; d < Dd; ++d) acc += b_in[d] * Bm[s * Dd + d];
    cvec[s] = acc;
  }
  // AT = A^T ; Apow[t] = (A^T)^(t+1)
  for (int idx = tid; idx < Ss * Ss; idx += 256) {
    int i = idx >> 6, j = idx & 63;
    float v = A[j * Ss + i];
    ATl[idx] = v;
    pa[idx] = v;
    Apow[idx] = v;
  }
  __syncthreads();
  float* prev = pa;
  float* next = pb;
  for (int t = 1; t < TC; ++t) {
    for (int idx = tid; idx < Ss * Ss; idx += 256) {
      int i = idx >> 6, j = idx & 63;
      float acc = 0.f;
      for (int k = 0; k < Ss; ++k) acc += prev[i * Ss + k] * ATl[k * Ss + j];
      next[idx] = acc;
      Apow[t * Ss * Ss + idx] = acc;
    }
    __syncthreads();
    float* tmp = prev; prev = next; next = tmp;
  }
}

// ---------------- K2: Bu[l][b][s] = x[b][l][:] @ M + cvec  (WMMA fp32) ------
__global__ __launch_bounds__(128) void k2_bu(const float* __restrict__ x,
                                             const float* __restrict__ M,
                                             const float* __restrict__ cvec,
                                             float* __restrict__ Bu) {
  __shared__ __align__(16) float xs[16 * Fp];
  __shared__ __align__(16) float Ms[Fp * Ss];
  int l = blockIdx.x;
  int b0 = blockIdx.y * 16;
  int tid = threadIdx.x;
  // async-stage M (32x64) into LDS while loading x tile through VGPRs
  for (int q = tid; q < (Fp * Ss) / 4; q += 128) async_cp16(M + q * 4, Ms + q * 4);
  for (int idx = tid; idx < 16 * Fp; idx += 128) {
    int r = idx / Fp, f = idx % Fp;
    xs[idx] = (f < Ff) ? x[((size_t)(b0 + r) * Ll + l) * Ff + f] : 0.f;
  }
  wait_async0();
  __syncthreads();
  int wave = tid >> 5, lane = tid & 31;
  int n0 = wave * 16;
  int r = lane & 15, h = lane >> 4;
  float cv = cvec[n0 + r];
  v8f acc;
#pragma unroll
  for (int v = 0; v < 8; ++v) acc[v] = cv;
#pragma unroll
  for (int kk = 0; kk < Fp / 4; ++kk) {
    v2f a = ldA(xs, Fp, 0, kk * 4, lane);
    v2f b = ldB(Ms, Ss, kk * 4, n0, lane);
    acc = wmma4(a, b, acc);
  }
#pragma unroll
  for (int v = 0; v < 8; ++v)
    Bu[((size_t)l * Bb + b0 + v + 8 * h) * Ss + n0 + r] = acc[v];
}

// ---------------- K3a: per-chunk local scan (in-place Bu -> local states) ---
__global__ __launch_bounds__(512) void k3a_scan(float* __restrict__ Bu,
                                                const float* __restrict__ Apow,
                                                float* __restrict__ Eend) {
  __shared__ __align__(16) float ATs[Ss * Ss];
  __shared__ __align__(16) float st[Ss * Ss];
  int tid = threadIdx.x, wave = tid >> 5, lane = tid & 31;
  int i0 = (wave >> 2) * 16, j0 = (wave & 3) * 16;
  int r = lane & 15, h = lane >> 4;
  int ch = blockIdx.x;
  for (int q = tid; q < (Ss * Ss) / 4; q += 512) async_cp16(Apow + q * 4, ATs + q * 4);
  v8f s;
#pragma unroll
  for (int v = 0; v < 8; ++v) s[v] = 0.f;
  wait_async0();
  __syncthreads();
  for (int t = 0; t < TC; ++t) {
    size_t l = (size_t)ch * TC + t;
#pragma unroll
    for (int v = 0; v < 8; ++v) st[(i0 + v + 8 * h) * Ss + j0 + r] = s[v];
    if (t + 1 < TC)
      __builtin_prefetch(&Bu[((l + 1) * Bb + i0 + 8 * h) * Ss + j0 + r], 0, 1);
    __syncthreads();
    v8f acc;
#pragma unroll
    for (int v = 0; v < 8; ++v)
      acc[v] = Bu[(l * Bb + i0 + v + 8 * h) * Ss + j0 + r];
#pragma unroll
    for (int k = 0; k < 16; ++k) {
      v2f a = ldA(st, Ss, i0, 4 * k, lane);
      v2f b = ldB(ATs, Ss, 4 * k, j0, lane);
      acc = wmma4(a, b, acc);
    }
    s = acc;
#pragma unroll
    for (int v = 0; v < 8; ++v)
      Bu[(l * Bb + i0 + v + 8 * h) * Ss + j0 + r] = s[v];  // local state in place
    __syncthreads();
  }
#pragma unroll
  for (int v = 0; v < 8; ++v)
    Eend[((size_t)ch * Bb + i0 + v + 8 * h) * Ss + j0 + r] = s[v];
}

// ---------------- K3b: sequential carry propagation over 64 chunks ----------
__global__ __launch_bounds__(512) void k3b_carry(const float* __restrict__ Eend,
                                                 const float* __restrict__ Apow,
                                                 float* __restrict__ Carry) {
  __shared__ __align__(16) float Ps[Ss * Ss];
  __shared__ __align__(16) float cs[Ss * Ss];
  int tid = threadIdx.x, wave = tid >> 5, lane = tid & 31;
  int i0 = (wave >> 2) * 16, j0 = (wave & 3) * 16;
  int r = lane & 15, h = lane >> 4;
  for (int q = tid; q < (Ss * Ss) / 4; q += 512)
    async_cp16(Apow + (TC - 1) * Ss * Ss + q * 4, Ps + q * 4);  // (A^T)^TC
  v8f c;
#pragma unroll
  for (int v = 0; v < 8; ++v) c[v] = 0.f;
#pragma unroll
  for (int v = 0; v < 8; ++v)
    Carry[(size_t)(i0 + v + 8 * h) * Ss + j0 + r] = 0.f;  // carry[chunk 0] = 0
  wait_async0();
  __syncthreads();
  for (int k2 = 0; k2 < NCH - 1; ++k2) {
#pragma unroll
    for (int v = 0; v < 8; ++v) cs[(i0 + v + 8 * h) * Ss + j0 + r] = c[v];
    __syncthreads();
    v8f acc;
#pragma unroll
    for (int v = 0; v < 8; ++v)
      acc[v] = Eend[((size_t)k2 * Bb + i0 + v + 8 * h) * Ss + j0 + r];
#pragma unroll
    for (int k = 0; k < 16; ++k) {
      v2f a = ldA(cs, Ss, i0, 4 * k, lane);
      v2f b = ldB(Ps, Ss, 4 * k, j0, lane);
      acc = wmma4(a, b, acc);
    }
    c = acc;
#pragma unroll
    for (int v = 0; v < 8; ++v)
      Carry[((size_t)(k2 + 1) * Bb + i0 + v + 8 * h) * Ss + j0 + r] = c[v];
    __syncthreads();
  }
}

// ------- K3c: carry correction + output proj + GELU + LN + pooled accum -----
// (A^T)^p tiles are async-prefetched into LDS one step ahead, hidden behind
// the output GEMM + GELU/LN phases.
__global__ __launch_bounds__(512) void k3c_out(const float* __restrict__ Sloc,
                                               const float* __restrict__ Carry,
                                               const float* __restrict__ Apow,
                                               const float* __restrict__ Cm,
                                               const float* __restrict__ ln_g,
                                               const float* __restrict__ ln_b,
                                               float* __restrict__ pooled) {
  __shared__ __align__(16) float smem[8192 + 8192 + 4096 + 4096 + 4096 + 8192 + 1024];
  float* CmT = smem;           // [64 x 128] Cm^T
  float* pool = smem + 8192;   // [64 x 128] block-local pooled accum
  float* cy = smem + 16384;    // [64 x 64] carry for this chunk
  float* sl = smem + 20480;    // [64 x 64] corrected state
  float* ap = smem + 24576;    // [64 x 64] (A^T)^(toff+1), async-prefetched
  float* yb = smem + 28672;    // [64 x 128] y staging
  float* red = smem + 36864;   // [64 x 16] row reduction scratch
  int tid = threadIdx.x, wave = tid >> 5, lane = tid & 31;
  int r = lane & 15, h = lane >> 4;
  int i0 = (wave >> 2) * 16, j0 = (wave & 3) * 16;  // correction tile
  int jd = (wave & 3) * 32;                         // output: two 16-wide tiles
  int l0 = blockIdx.x * LPB;
  int ch = l0 / TC;
  for (int idx = tid; idx < Ss * Dd; idx += 512) {
    int s = idx / Dd, d = idx % Dd;
    CmT[idx] = Cm[d * Ss + s];
    pool[idx] = 0.f;
  }
  for (int idx = tid; idx < Ss * Ss; idx += 512)
    cy[idx] = Carry[(size_t)ch * Ss * Ss + idx];
  // async-prefetch Apow tile for t = 0
  {
    int toff0 = l0 % TC;
    for (int q = tid; q < (Ss * Ss) / 4; q += 512)
      async_cp16(Apow + (size_t)toff0 * Ss * Ss + q * 4, ap + q * 4);
  }
  wait_async0();
  __syncthreads();
  int row = tid & 63, g = tid >> 6;  // LN mapping: 8 col-groups of 16 per row
  for (int t = 0; t < LPB; ++t) {
    size_t l = (size_t)l0 + t;
    // corrected state: s_l = local_l + carry @ Apow[toff]
    v8f acc;
#pragma unroll
    for (int v = 0; v < 8; ++v)
      acc[v] = Sloc[(l * Bb + i0 + v + 8 * h) * Ss + j0 + r];
#pragma unroll
    for (int k = 0; k < 16; ++k) {
      v2f a = ldA(cy, Ss, i0, 4 * k, lane);
      v2f b = ldB(ap, Ss, 4 * k, j0, lane);
      acc = wmma4(a, b, acc);
    }
#pragma unroll
    for (int v = 0; v < 8; ++v) sl[(i0 + v + 8 * h) * Ss + j0 + r] = acc[v];
    __syncthreads();  // all waves done reading ap; sl ready
    // kick off next step's Apow tile (overlaps output GEMM + GELU/LN)
    if (t + 1 < LPB) {
      int toffn = (int)((l + 1) % TC);
      for (int q = tid; q < (Ss * Ss) / 4; q += 512)
        async_cp16(Apow + (size_t)toffn * Ss * Ss + q * 4, ap + q * 4);
    }
    // output projection: y(64x128) = s_l(64x64) @ CmT(64x128)
    v8f y0, y1;
#pragma unroll
    for (int v = 0; v < 8; ++v) { y0[v] = 0.f; y1[v] = 0.f; }
#pragma unroll
    for (int k = 0; k < 16; ++k) {
      v2f a = ldA(sl, Ss, i0, 4 * k, lane);
      v2f b0 = ldB(CmT, Dd, 4 * k, jd, lane);
      v2f b1 = ldB(CmT, Dd, 4 * k, jd + 16, lane);
      y0 = wmma4(a, b0, y0);
      y1 = wmma4(a, b1, y1);
    }
#pragma unroll
    for (int v = 0; v < 8; ++v) {
      yb[(i0 + v + 8 * h) * Dd + jd + r] = y0[v];
      yb[(i0 + v + 8 * h) * Dd + jd + 16 + r] = y1[v];
    }
    __syncthreads();
    // GELU + partial row sums
    float sm = 0.f, s2 = 0.f;
    for (int c2 = 0; c2 < 16; ++c2) {
      int d = g * 16 + c2;
      float v = gelu_exact(yb[row * Dd + d]);
      yb[row * Dd + d] = v;
      sm += v;
      s2 += v * v;
    }
    red[row * 16 + g * 2 + 0] = sm;
    red[row * 16 + g * 2 + 1] = s2;
    __syncthreads();
    if (tid < 64) {
      float S1 = 0.f, S2 = 0.f;
      for (int g2 = 0; g2 < 8; ++g2) {
        S1 += red[tid * 16 + g2 * 2 + 0];
        S2 += red[tid * 16 + g2 * 2 + 1];
      }
      float mean = S1 * (1.0f / Dd);
      float var = S2 * (1.0f / Dd) - mean * mean;
      red[tid * 16 + 0] = mean;
      red[tid * 16 + 1] = rsqrtf(var + 1e-5f);
    }
    __syncthreads();
    {
      float mean = red[row * 16 + 0], inv = red[row * 16 + 1];
      for (int c2 = 0; c2 < 16; ++c2) {
        int d = g * 16 + c2;
        float v = (yb[row * Dd + d] - mean) * inv * ln_g[d] + ln_b[d];
        pool[row * Dd + d] += v * (1.0f / Ll);
      }
    }
    wait_async0();   // next ap tile landed
    __syncthreads();
  }
  for (int idx = tid; idx < Bb * Dd; idx += 512) atomicAdd(&pooled[idx], pool[idx]);
}

// ---------------- K4: logits = pooled @ W_fc^T + b_fc -----------------------
__global__ __launch_bounds__(640) void k4_fc(const float* __restrict__ pooled,
                                             const float* __restrict__ W_fc,
                                             const float* __restrict__ b_fc,
                                             float* __restrict__ out) {
  int tid = threadIdx.x;
  if (tid < Bb * Cc) {
    int b = tid / Cc, c = tid % Cc;
    float acc = b_fc[c];
    for (int d = 0; d < Dd; ++d) acc += pooled[b * Dd + d] * W_fc[c * Dd + d];
    out[b * Cc + c] = acc;
  }
}

extern "C" void kernel_launch(void* const* d_in, const int* in_sizes, int n_in,
                              void* d_out, int out_size, void* d_ws, size_t ws_size,
                              hipStream_t stream) {
  const float* x = (const float*)d_in[0];
  const float* W_in = (const float*)d_in[1];
  const float* b_in = (const float*)d_in[2];
  const float* A = (const float*)d_in[3];
  const float* Bm = (const float*)d_in[4];
  const float* Cm = (const float*)d_in[5];
  const float* ln_g = (const float*)d_in[6];
  const float* ln_b = (const float*)d_in[7];
  const float* W_fc = (const float*)d_in[8];
  const float* b_fc = (const float*)d_in[9];
  float* out = (float*)d_out;

  float* ws = (float*)d_ws;
  float* Bu = ws;                               // L*B*S (becomes states in place)
  float* Mm = Bu + (size_t)Ll * Bb * Ss;        // Fp*Ss
  float* cvec = Mm + Fp * Ss;                   // Ss
  float* Apow = cvec + Ss;                      // TC*Ss*Ss
  float* Eend = Apow + TC * Ss * Ss;            // NCH*Ss*Ss
  float* Carry = Eend + NCH * Ss * Ss;          // NCH*Ss*Ss
  float* pooled = Carry + NCH * Ss * Ss;        // Bb*Dd

  k1_prep<<<1, 256, 0, stream>>>(W_in, b_in, A, Bm, Mm, cvec, Apow);
  k2_bu<<<dim3(Ll, Bb / 16), 128, 0, stream>>>(x, Mm, cvec, Bu);
  k3a_scan<<<NCH, 512, 0, stream>>>(Bu, Apow, Eend);
  k3b_carry<<<1, 512, 0, stream>>>(Eend, Apow, Carry);
  (void)hipMemsetAsync(pooled, 0, Bb * Dd * sizeof(float), stream);
  k3c_out<<<Ll / LPB, 512, 0, stream>>>(Bu, Carry, Apow, Cm, ln_g, ln_b, pooled);
  k4_fc<<<1, 640, 0, stream>>>(pooled, W_fc, b_fc, out);
}